// GATConv_dgl_32126355374977
// MI455X (gfx1250) — compile-verified
//
#include <hip/hip_runtime.h>
#include <hip/hip_bf16.h>

// ---------------------------------------------------------------------------
// GATConv for MI455X (gfx1250, wave32).
// Phase 1: feat = x @ fc_w via V_WMMA_F32_16X16X4_F32 (fp32 WMMA, memory-bound
//          GEMM so fp32 precision is free).
// Phase 2: el/er per-node attention logits (wave reductions).
// Phase 3: 3 edge passes (segment-max via u32-key atomicMax, exp-sum via
//          atomicAdd, aggregation with head-mean folded in -> 4x fewer atomics,
//          output accumulator is L2-resident).
// ---------------------------------------------------------------------------

typedef float v2f __attribute__((ext_vector_type(2)));
typedef float v8f __attribute__((ext_vector_type(8)));

#define GAT_HEADS   4
#define GAT_D       32
#define GAT_FDIM    128   // HEADS * D == IN_DIM
#define NEG_SLOPE   0.2f

// ---------------- Phase 1: WMMA GEMM  feat[N,128] = x[N,128] @ w[128,128] ---
// One wave computes a 16-row x 128-col strip: 8 accumulators (16x16 tiles),
// K consumed 4 at a time by v_wmma_f32_16x16x4_f32.
__global__ __launch_bounds__(128) void gat_gemm_feat(
    const float* __restrict__ x, const float* __restrict__ w,
    float* __restrict__ feat, int nNodes)
{
    const int lane  = threadIdx.x & 31;
    const int wave  = threadIdx.x >> 5;
    const int mTile = blockIdx.x * 4 + wave;
    const int mTiles = (nNodes + 15) >> 4;
    if (mTile >= mTiles) return;

    const int laneM  = lane & 15;        // row within tile (B/C column index)
    const int khalf  = lane >> 4;        // 0: K pair {k,k+1}, 1: {k+2,k+3}
    int row = mTile * 16 + laneM;
    int rowClamped = row < nNodes ? row : nNodes - 1;   // tail-safe loads

    v8f acc[8];
    #pragma unroll
    for (int t = 0; t < 8; ++t) acc[t] = v8f{};

    const float* xrow = x + (size_t)rowClamped * GAT_FDIM;

    for (int k0 = 0; k0 < GAT_FDIM; k0 += 4) {
        // A fragment: 16x4 f32. Lane holds K = k0+2*khalf, k0+2*khalf+1.
        v2f a = *(const v2f*)(xrow + k0 + 2 * khalf);
        const int kb = k0 + 2 * khalf;
        #pragma unroll
        for (int t = 0; t < 8; ++t) {
            // B fragment: 4x16 f32 tile at columns [t*16, t*16+16).
            const int ncol = t * 16 + laneM;
            v2f b;
            b.x = w[(kb + 0) * GAT_FDIM + ncol];
            b.y = w[(kb + 1) * GAT_FDIM + ncol];
            acc[t] = __builtin_amdgcn_wmma_f32_16x16x4_f32(
                /*neg_a=*/false, a, /*neg_b=*/false, b,
                /*c_mod=*/(short)0, acc[t],
                /*reuse_a=*/false, /*reuse_b=*/false);
        }
    }

    // C/D layout: VGPR r holds M = r + 8*(lane>=16), N = lane&15.
    const int mOff = 8 * khalf;
    #pragma unroll
    for (int t = 0; t < 8; ++t) {
        #pragma unroll
        for (int r = 0; r < 8; ++r) {
            int m = mTile * 16 + mOff + r;
            if (m < nNodes)
                feat[(size_t)m * GAT_FDIM + t * 16 + laneM] = acc[t][r];
        }
    }
}

// ---------------- Phase 2: el/er logits ------------------------------------
// Block = 128 threads = 4 waves; wave h reduces head h of one node.
__global__ __launch_bounds__(128) void gat_node_attn(
    const float* __restrict__ feat,
    const float* __restrict__ attn_l, const float* __restrict__ attn_r,
    float* __restrict__ el, float* __restrict__ er, int nNodes)
{
    const int n = blockIdx.x;
    if (n >= nNodes) return;
    const int tid = threadIdx.x;
    const int h = tid >> 5, d = tid & 31;

    float v  = feat[(size_t)n * GAT_FDIM + tid];
    float pl = v * attn_l[h * GAT_D + d];
    float pr = v * attn_r[h * GAT_D + d];
    #pragma unroll
    for (int off = 16; off > 0; off >>= 1) {
        pl += __shfl_down(pl, off, 32);
        pr += __shfl_down(pr, off, 32);
    }
    if (d == 0) {
        el[n * GAT_HEADS + h] = pl;
        er[n * GAT_HEADS + h] = pr;
    }
}

// ---------------- float <-> monotone-u32 key for atomic max ----------------
__device__ __forceinline__ unsigned fkey(float f) {
    unsigned u = __float_as_uint(f);
    return (u & 0x80000000u) ? ~u : (u | 0x80000000u);
}
__device__ __forceinline__ float funkey(unsigned k) {
    return (k & 0x80000000u) ? __uint_as_float(k ^ 0x80000000u)
                             : __uint_as_float(~k);
}
__device__ __forceinline__ float lrelu(float v) {
    return v > 0.f ? v : NEG_SLOPE * v;
}

// ---------------- Init: out = mean_h(bias), m = -inf key, s = 0 ------------
__global__ void gat_init(float* __restrict__ out, unsigned* __restrict__ mkey,
                         float* __restrict__ s, const float* __restrict__ bias,
                         int nNodes)
{
    int i = blockIdx.x * blockDim.x + threadIdx.x;
    if (i < nNodes * GAT_D) {
        int d = i & (GAT_D - 1);
        float b = 0.f;
        #pragma unroll
        for (int h = 0; h < GAT_HEADS; ++h) b += bias[h * GAT_D + d];
        out[i] = b * (1.f / GAT_HEADS);
    }
    if (i < nNodes * GAT_HEADS) { mkey[i] = 0u; s[i] = 0.f; }
}

// ---------------- Edge pass A: segment max ---------------------------------
__global__ void gat_edge_max(
    const int* __restrict__ src, const int* __restrict__ dst,
    const float* __restrict__ el, const float* __restrict__ er,
    unsigned* __restrict__ mkey, int nEdges)
{
    int i = blockIdx.x * blockDim.x + threadIdx.x;
    if (i >= nEdges * GAT_HEADS) return;
    int e = i >> 2, h = i & 3;
    int u = src[e], v = dst[e];
    float ev = lrelu(el[u * GAT_HEADS + h] + er[v * GAT_HEADS + h]);
    atomicMax(&mkey[v * GAT_HEADS + h], fkey(ev));
}

// ---------------- Edge pass B: segment sum of exp(e - m) -------------------
__global__ void gat_edge_sum(
    const int* __restrict__ src, const int* __restrict__ dst,
    const float* __restrict__ el, const float* __restrict__ er,
    const unsigned* __restrict__ mkey, float* __restrict__ s, int nEdges)
{
    int i = blockIdx.x * blockDim.x + threadIdx.x;
    if (i >= nEdges * GAT_HEADS) return;
    int e = i >> 2, h = i & 3;
    int u = src[e], v = dst[e];
    float ev = lrelu(el[u * GAT_HEADS + h] + er[v * GAT_HEADS + h]);
    float m  = funkey(mkey[v * GAT_HEADS + h]);
    atomicAdd(&s[v * GAT_HEADS + h], __expf(ev - m));
}

// ---------------- Edge pass C: aggregate with folded head-mean -------------
// One wave per edge; lane d accumulates 0.25 * sum_h alpha_h * feat[u,h,d]
// with a single atomicAdd into the final [N,32] output (L2-resident).
__global__ __launch_bounds__(256) void gat_edge_aggr(
    const int* __restrict__ src, const int* __restrict__ dst,
    const float* __restrict__ el, const float* __restrict__ er,
    const unsigned* __restrict__ mkey, const float* __restrict__ s,
    const float* __restrict__ feat, float* __restrict__ out, int nEdges)
{
    const int lane = threadIdx.x & 31;
    const int e = (blockIdx.x * blockDim.x + threadIdx.x) >> 5;
    if (e >= nEdges) return;
    const int u = src[e], v = dst[e];

    float alpha = 0.f;
    if (lane < GAT_HEADS) {
        const int h = lane;
        float ev = lrelu(el[u * GAT_HEADS + h] + er[v * GAT_HEADS + h]);
        float m  = funkey(mkey[v * GAT_HEADS + h]);
        float sv = s[v * GAT_HEADS + h];
        alpha = __expf(ev - m) / fmaxf(sv, 1e-16f);
    }
    const float a0 = __shfl(alpha, 0, 32);
    const float a1 = __shfl(alpha, 1, 32);
    const float a2 = __shfl(alpha, 2, 32);
    const float a3 = __shfl(alpha, 3, 32);

    const float* f = feat + (size_t)u * GAT_FDIM;
    float accv = a0 * f[lane] + a1 * f[GAT_D + lane] +
                 a2 * f[2 * GAT_D + lane] + a3 * f[3 * GAT_D + lane];
    atomicAdd(out + (size_t)v * GAT_D + lane, 0.25f * accv);
}

// ---------------------------------------------------------------------------
extern "C" void kernel_launch(void* const* d_in, const int* in_sizes, int n_in,
                              void* d_out, int out_size, void* d_ws, size_t ws_size,
                              hipStream_t stream)
{
    const float* x        = (const float*)d_in[0];
    const int*   edge_src = (const int*)  d_in[1];
    const int*   edge_dst = (const int*)  d_in[2];
    const float* fc_w     = (const float*)d_in[3];
    const float* attn_l   = (const float*)d_in[4];
    const float* attn_r   = (const float*)d_in[5];
    const float* bias     = (const float*)d_in[6];
    float*       out      = (float*)d_out;

    const int nNodes = in_sizes[0] / GAT_FDIM;
    const int nEdges = in_sizes[1];

    // Workspace layout (floats): feat | el | er | mkey(u32) | s
    float*    feat = (float*)d_ws;
    float*    el   = feat + (size_t)nNodes * GAT_FDIM;
    float*    er   = el   + (size_t)nNodes * GAT_HEADS;
    unsigned* mkey = (unsigned*)(er + (size_t)nNodes * GAT_HEADS);
    float*    s    = (float*)(mkey + (size_t)nNodes * GAT_HEADS);

    // 1) feat = x @ fc_w (WMMA fp32)
    {
        int mTiles = (nNodes + 15) / 16;
        int blocks = (mTiles + 3) / 4;        // 4 waves / block, 1 M-tile / wave
        gat_gemm_feat<<<blocks, 128, 0, stream>>>(x, fc_w, feat, nNodes);
    }
    // 2) el / er
    gat_node_attn<<<nNodes, 128, 0, stream>>>(feat, attn_l, attn_r, el, er, nNodes);
    // 3) init output (head-mean of bias) + m/s accumulators
    {
        int total = nNodes * GAT_D;
        gat_init<<<(total + 255) / 256, 256, 0, stream>>>(out, mkey, s, bias, nNodes);
    }
    // 4) segment max
    {
        int total = nEdges * GAT_HEADS;
        gat_edge_max<<<(total + 255) / 256, 256, 0, stream>>>(
            edge_src, edge_dst, el, er, mkey, nEdges);
    }
    // 5) segment exp-sum
    {
        int total = nEdges * GAT_HEADS;
        gat_edge_sum<<<(total + 255) / 256, 256, 0, stream>>>(
            edge_src, edge_dst, el, er, mkey, s, nEdges);
    }
    // 6) aggregation (1 wave / edge), head-mean folded in
    {
        long long total = (long long)nEdges * 32;
        int blocks = (int)((total + 255) / 256);
        gat_edge_aggr<<<blocks, 256, 0, stream>>>(
            edge_src, edge_dst, el, er, mkey, s, feat, out, nEdges);
    }
}